// MultiheadAttention_13022340842322
// MI455X (gfx1250) — compile-verified
//
#include <hip/hip_runtime.h>
#include <math.h>

// ---------------------------------------------------------------------------
// Complex multi-head attention for MI455X (gfx1250), f32 WMMA path.
//   B=4, L=1024, F=1024, H=16, d=64. Matrix math via V_WMMA_F32_16X16X4_F32
//   (wave32, A=16x4 in 2 VGPRs, B=4x16 in 2 VGPRs, C/D=16x16 in 8 VGPRs).
// HBM-bound (~0.25 GB vs ~155 GFLOP) -> keep f32, never materialize the LxL
// complex attention matrix (16x1024 complex logit block lives in LDS), and
// use the CDNA5 Tensor Data Mover to DMA K tiles (strided rows) into LDS,
// double-buffered, tracked with TENSORcnt.
// ---------------------------------------------------------------------------

#define B_ 4
#define L_ 1024
#define F_ 1024
#define H_ 16
#define D_ 64
#define EPS_ 1e-6f

typedef __attribute__((ext_vector_type(2))) float v2f;
typedef __attribute__((ext_vector_type(8))) float v8f;
typedef unsigned int v4u __attribute__((ext_vector_type(4)));
typedef unsigned int v8u __attribute__((ext_vector_type(8)));

__device__ __forceinline__ v8f wmma4(v2f a, v2f b, v8f c) {
  // D = A(16x4) * B(4x16) + C, fp32.
  return __builtin_amdgcn_wmma_f32_16x16x4_f32(false, a, false, b, (short)0, c,
                                               false, false);
}

__device__ __forceinline__ v8f vzero8() {
  v8f z = {0.f, 0.f, 0.f, 0.f, 0.f, 0.f, 0.f, 0.f};
  return z;
}

// ---------------------------------------------------------------------------
// Tensor Data Mover: DMA a 16-row x 64-elem f32 tile (row pitch = stride
// elements) from global memory into LDS at byte offset lds_byte_off.
// D# per CDNA5 ISA ch.8: group0 = {count, lds_addr, global_addr[56:0], type=2},
// group1 = {mask/data_size, tensor_dim0/1, tile_dim0/1/2, dim0_stride}.
// Tracked with TENSORcnt.
// ---------------------------------------------------------------------------
__device__ __forceinline__ void tdm_load_tile16x64(uint32_t lds_byte_off,
                                                   const float* gptr,
                                                   uint32_t stride_elems) {
  const unsigned long long ga = (unsigned long long)(uintptr_t)gptr;
  v4u g0;
  g0[0] = 1u;                                    // count=1, user descriptor
  g0[1] = lds_byte_off;                          // lds_addr (bytes)
  g0[2] = (uint32_t)ga;                          // global_addr[31:0]
  g0[3] = ((uint32_t)(ga >> 32) & 0x01FFFFFFu)   // global_addr[56:32]
          | 0x80000000u;                         // type=2 ("image")
  v8u g1;
  g1[0] = 0x00020000u;        // workgroup_mask=0 (no cluster), data_size=4B
  g1[1] = 64u << 16;          // tensor_dim0 = 64      (bits 79:48, low half)
  g1[2] = 16u << 16;          // tensor_dim0 hi=0 | tensor_dim1 = 16
  g1[3] = 64u << 16;          // tensor_dim1 hi=0 | tile_dim0 = 64
  g1[4] = 16u;                // tile_dim1 = 16 | tile_dim2 = 0 (2D tile)
  g1[5] = stride_elems;       // tensor_dim0_stride[31:0] (row pitch)
  g1[6] = 0u;                 // stride hi | tensor_dim1_stride lo (unused, 2D)
  g1[7] = 0u;
  asm volatile("tensor_load_to_lds %0, %1" ::"s"(g0), "s"(g1) : "memory");
}

__device__ __forceinline__ void wait_tensorcnt(int n) {
  if (n == 0)
    asm volatile("s_wait_tensorcnt 0x0" ::: "memory");
  else
    asm volatile("s_wait_tensorcnt 0x2" ::: "memory");
}

// ---------------------------------------------------------------------------
// Kernel 1: Y = LayerNorm_c(X @ W), X real [M,F], W complex [F,F].
// Block = 256 threads (8 waves) owns 16 rows x all 1024 columns so the complex
// LayerNorm (reduce over features) fuses into the GEMM epilogue.
// ---------------------------------------------------------------------------
__global__ __launch_bounds__(256) void proj_ln_kernel(
    const float* __restrict__ X, const float2* __restrict__ W,
    const float* __restrict__ scale, const float* __restrict__ bias,
    float* __restrict__ outR, float* __restrict__ outI) {
  __shared__ float s_sr[16], s_si[16], s_s2[16];

  const int lane = threadIdx.x & 31;
  const int wv = threadIdx.x >> 5;
  const int half = lane >> 4;
  const int lr = lane & 15;
  const int rb = blockIdx.x * 16;

  if (threadIdx.x < 16) {
    s_sr[threadIdx.x] = 0.f;
    s_si[threadIdx.x] = 0.f;
    s_s2[threadIdx.x] = 0.f;
  }
  __syncthreads();

  v8f accR[8], accI[8];
#pragma unroll
  for (int i = 0; i < 8; ++i) {
    accR[i] = vzero8();
    accI[i] = vzero8();
  }

  const float* Arow = X + (size_t)(rb + lr) * F_;
  for (int k = 0; k < F_; k += 4) {
    v2f a = *(const v2f*)(Arow + k + 2 * half);
    __builtin_prefetch(Arow + k + 32, 0, 3);  // global_prefetch_b8
#pragma unroll
    for (int i = 0; i < 8; ++i) {
      const int n = (wv * 8 + i) * 16 + lr;
      const float2 w0 = W[(size_t)(k + 2 * half) * F_ + n];
      const float2 w1 = W[(size_t)(k + 2 * half + 1) * F_ + n];
      v2f br = {w0.x, w1.x};
      v2f bi = {w0.y, w1.y};
      accR[i] = wmma4(a, br, accR[i]);
      accI[i] = wmma4(a, bi, accI[i]);
    }
  }

  // ---- fused complex LayerNorm reductions ---------------------------------
#pragma unroll
  for (int j = 0; j < 8; ++j) {
    float sr = 0.f, si = 0.f, s2 = 0.f;
#pragma unroll
    for (int i = 0; i < 8; ++i) {
      const float xr = accR[i][j], xi = accI[i][j];
      sr += xr;
      si += xi;
      s2 += xr * xr + xi * xi;
    }
#pragma unroll
    for (int m = 1; m < 16; m <<= 1) {
      sr += __shfl_xor(sr, m, 32);
      si += __shfl_xor(si, m, 32);
      s2 += __shfl_xor(s2, m, 32);
    }
    if (lr == 0) {
      const int r = j + 8 * half;
      atomicAdd(&s_sr[r], sr);
      atomicAdd(&s_si[r], si);
      atomicAdd(&s_s2[r], s2);
    }
  }
  __syncthreads();

#pragma unroll
  for (int j = 0; j < 8; ++j) {
    const int r = j + 8 * half;
    const float mr = s_sr[r] * (1.0f / F_);
    const float mi = s_si[r] * (1.0f / F_);
    const float var = s_s2[r] * (1.0f / F_) - (mr * mr + mi * mi);
    const float inv = rsqrtf(var + EPS_);
    const size_t orow = (size_t)(rb + r) * F_;
#pragma unroll
    for (int i = 0; i < 8; ++i) {
      const int n = (wv * 8 + i) * 16 + lr;
      const float sc = scale[n] * inv;
      outR[orow + n] = (accR[i][j] - mr) * sc + bias[n];  // bias -> real part
      outI[orow + n] = (accI[i][j] - mi) * sc;
    }
  }
}

// ---------------------------------------------------------------------------
// Kernel 2: fused attention per (b, h, 16-query tile).
// Stage 1: S = Q Kᵀ / 8 (complex) -> LDS; K tiles DMAed by the TDM (double
//          buffered per wave, TENSORcnt-synchronized).
// Stage 2: lexicographic max, complex exp/sum/divide in LDS.
// Stage 3: O = P V (complex), cross-wave reduce via ds_add_f32.
// LDS: 2*16*1024 logits + reductions + O accum + 8 waves * 2 buf * 2 planes
//      * 4KB staging = 272,512 bytes (needs CDNA5's 320KB WGP LDS).
// ---------------------------------------------------------------------------
#define ATTN_LDS_BASE (2 * 16 * L_ + 2 * 256 + 2 * 16 + 2 * 16 * 64)  // 35360
#define ATTN_LDS_FLOATS (ATTN_LDS_BASE + 8 * 4096)                    // 68128

__global__ __launch_bounds__(256) void attn_kernel(
    const float* __restrict__ Qr, const float* __restrict__ Qi,
    const float* __restrict__ Kr, const float* __restrict__ Ki,
    const float* __restrict__ Vr, const float* __restrict__ Vi,
    const float* __restrict__ mask, float* __restrict__ OutR,
    float* __restrict__ OutI) {
  extern __shared__ float lds[];
  float* Sr = lds;                  // 16*1024
  float* Si = Sr + 16 * L_;         // 16*1024
  float* redA = Si + 16 * L_;       // 16*16
  float* redB = redA + 256;         // 16*16
  float* rowA = redB + 256;         // 16
  float* rowB = rowA + 16;          // 16
  float* Oar = rowB + 16;           // 16*64
  float* Oai = Oar + 16 * 64;       // 16*64
  float* KStage = lds + ATTN_LDS_BASE;  // per-wave: 2 buf x (Kr 1024|Ki 1024)

  const int qt = blockIdx.x & 63;
  const int h = (blockIdx.x >> 6) & 15;
  const int b = blockIdx.x >> 10;
  const int qb = qt * 16;
  const int hoff = h * D_;

  const int lane = threadIdx.x & 31;
  const int wv = threadIdx.x >> 5;
  const int half = lane >> 4;
  const int lr = lane & 15;

  // ---- Stage 1: complex logits -------------------------------------------
  // Hoist all 16 d-step A-operands for Q (re+im) into registers once.
  const size_t qrow = (size_t)(b * L_ + qb + lr) * F_ + hoff;
  v2f qRe[16], qIm[16];
#pragma unroll
  for (int dd = 0; dd < 16; ++dd) {
    qRe[dd] = *(const v2f*)(Qr + qrow + dd * 4 + 2 * half);
    qIm[dd] = *(const v2f*)(Qi + qrow + dd * 4 + 2 * half);
  }

  const int wvu = __builtin_amdgcn_readfirstlane(wv);  // wave-uniform in SGPR
  float* stage = KStage + wvu * 4096;
  const size_t khead = (size_t)(b * L_) * F_ + hoff;

  // TDM prologue: DMA key tile 0 (re + im planes) into buffer 0.
  {
    const size_t g = khead + (size_t)((wvu * 8) * 16) * F_;
    tdm_load_tile16x64((uint32_t)(uintptr_t)(stage), Kr + g, F_);
    tdm_load_tile16x64((uint32_t)(uintptr_t)(stage + 1024), Ki + g, F_);
  }

  for (int t = 0; t < 8; ++t) {
    const int kb = (wvu * 8 + t) * 16;
    float* buf = stage + (t & 1) * 2048;
    if (t < 7) {  // DMA next tile into the other buffer, then wait for current
      float* nbuf = stage + ((t + 1) & 1) * 2048;
      const size_t g = khead + (size_t)((wvu * 8 + t + 1) * 16) * F_;
      tdm_load_tile16x64((uint32_t)(uintptr_t)(nbuf), Kr + g, F_);
      tdm_load_tile16x64((uint32_t)(uintptr_t)(nbuf + 1024), Ki + g, F_);
      wait_tensorcnt(2);  // only the 2 newest (prefetch) may be outstanding
    } else {
      wait_tensorcnt(0);
    }

    const float* KrS = buf;          // tile row-major: [token 0..15][d 0..63]
    const float* KiS = buf + 1024;
    v8f sR = vzero8(), sI = vzero8();
#pragma unroll
    for (int dd = 0; dd < 16; ++dd) {
      const int d = dd * 4;
      v2f bR = *(const v2f*)(KrS + lr * 64 + d + 2 * half);
      v2f bI = *(const v2f*)(KiS + lr * 64 + d + 2 * half);
      v2f aIn = -qIm[dd];  // f32 WMMA has no A-neg bit; negate in VALU
      sR = wmma4(qRe[dd], bR, sR);
      sR = wmma4(aIn, bI, sR);
      sI = wmma4(qRe[dd], bI, sI);
      sI = wmma4(qIm[dd], bR, sI);
    }
#pragma unroll
    for (int j = 0; j < 8; ++j) {
      const int m = j + 8 * half;
      const int kc = kb + lr;
      const float mv = mask[((size_t)b * L_ + (qb + m)) * L_ + kc];
      float re = sR[j] * 0.125f;  // 1/sqrt(64)
      float im = sI[j] * 0.125f;
      if (mv <= 0.f) {
        re = -__builtin_inff();
        im = 0.f;
      }
      Sr[m * L_ + kc] = re;
      Si[m * L_ + kc] = im;
    }
  }
  __syncthreads();

  // ---- Stage 2: complex softmax over k (in LDS) ---------------------------
  const int r = threadIdx.x & 15;  // row
  const int c = threadIdx.x >> 4;  // 64-wide chunk 0..15
  {
    float mr = -__builtin_inff(), mi = -__builtin_inff();
    for (int k = c * 64; k < c * 64 + 64; ++k) {
      const float sr = Sr[r * L_ + k], si = Si[r * L_ + k];
      if (sr > mr || (sr == mr && si > mi)) {
        mr = sr;
        mi = si;
      }
    }
    redA[r * 16 + c] = mr;
    redB[r * 16 + c] = mi;
  }
  __syncthreads();
  if (threadIdx.x < 16) {
    float mr = -__builtin_inff(), mi = -__builtin_inff();
    for (int cc = 0; cc < 16; ++cc) {
      const float sr = redA[threadIdx.x * 16 + cc];
      const float si = redB[threadIdx.x * 16 + cc];
      if (sr > mr || (sr == mr && si > mi)) {
        mr = sr;
        mi = si;
      }
    }
    rowA[threadIdx.x] = mr;
    rowB[threadIdx.x] = mi;
  }
  __syncthreads();
  {
    const float mr = rowA[r], mi = rowB[r];
    float sumr = 0.f, sumi = 0.f;
    for (int k = c * 64; k < c * 64 + 64; ++k) {
      const float sr = Sr[r * L_ + k], si = Si[r * L_ + k];
      const float er = __expf(sr - mr);  // exp(z) = e^re (cos im + i sin im)
      float sn, cs;
      __sincosf(si - mi, &sn, &cs);
      const float pr = er * cs, pi = er * sn;
      Sr[r * L_ + k] = pr;
      Si[r * L_ + k] = pi;
      sumr += pr;
      sumi += pi;
    }
    redA[r * 16 + c] = sumr;
    redB[r * 16 + c] = sumi;
  }
  __syncthreads();
  if (threadIdx.x < 16) {
    float sumr = 0.f, sumi = 0.f;
    for (int cc = 0; cc < 16; ++cc) {
      sumr += redA[threadIdx.x * 16 + cc];
      sumi += redB[threadIdx.x * 16 + cc];
    }
    const float den = sumr * sumr + sumi * sumi;
    rowA[threadIdx.x] = sumr / den;  // complex reciprocal of the sum
    rowB[threadIdx.x] = -sumi / den;
  }
  __syncthreads();
  {
    const float ir = rowA[r], ii = rowB[r];
    for (int k = c * 64; k < c * 64 + 64; ++k) {
      const float pr = Sr[r * L_ + k], pi = Si[r * L_ + k];
      Sr[r * L_ + k] = pr * ir - pi * ii;
      Si[r * L_ + k] = pr * ii + pi * ir;
    }
  }
  for (int i = threadIdx.x; i < 16 * 64; i += 256) {
    Oar[i] = 0.f;
    Oai[i] = 0.f;
  }
  __syncthreads();

  // ---- Stage 3: O = P @ V, each wave covers 128 k-tokens, all 4 d-tiles ---
  v8f oR[4], oI[4];
#pragma unroll
  for (int nt = 0; nt < 4; ++nt) {
    oR[nt] = vzero8();
    oI[nt] = vzero8();
  }
  for (int kk = wv * 128; kk < wv * 128 + 128; kk += 4) {
    v2f aR = *(const v2f*)(Sr + lr * L_ + kk + 2 * half);  // A from LDS
    v2f aI = *(const v2f*)(Si + lr * L_ + kk + 2 * half);
    v2f aIn = -aI;
#pragma unroll
    for (int nt = 0; nt < 4; ++nt) {
      const int n = nt * 16 + lr;
      const size_t v0 = (size_t)(b * L_ + kk + 2 * half) * F_ + hoff + n;
      const size_t v1 = (size_t)(b * L_ + kk + 2 * half + 1) * F_ + hoff + n;
      if (nt == 0) {  // prefetch 8 tokens ahead (unrolled: no branch)
        __builtin_prefetch(Vr + v0 + 8 * F_, 0, 2);
        __builtin_prefetch(Vi + v0 + 8 * F_, 0, 2);
      }
      v2f bR = {Vr[v0], Vr[v1]};
      v2f bI = {Vi[v0], Vi[v1]};
      oR[nt] = wmma4(aR, bR, oR[nt]);
      oR[nt] = wmma4(aIn, bI, oR[nt]);
      oI[nt] = wmma4(aR, bI, oI[nt]);
      oI[nt] = wmma4(aI, bR, oI[nt]);
    }
  }
#pragma unroll
  for (int nt = 0; nt < 4; ++nt)
#pragma unroll
    for (int j = 0; j < 8; ++j) {
      const int m = j + 8 * half;
      const int n = nt * 16 + lr;
      atomicAdd(&Oar[m * 64 + n], oR[nt][j]);  // ds_add_f32 cross-wave reduce
      atomicAdd(&Oai[m * 64 + n], oI[nt][j]);
    }
  __syncthreads();

  for (int i = threadIdx.x; i < 16 * 64; i += 256) {
    const int m = i >> 6, n = i & 63;
    const size_t o = (size_t)(b * L_ + qb + m) * F_ + hoff + n;
    OutR[o] = Oar[i];
    OutI[o] = Oai[i];
  }
}

// ---------------------------------------------------------------------------
// Kernel 3: final complex GEMM  out = O @ Wo, interleaved complex64 output.
// ---------------------------------------------------------------------------
__global__ __launch_bounds__(256) void oproj_kernel(
    const float* __restrict__ Ar, const float* __restrict__ Ai,
    const float2* __restrict__ W, float2* __restrict__ out) {
  const int lane = threadIdx.x & 31;
  const int wv = threadIdx.x >> 5;
  const int half = lane >> 4;
  const int lr = lane & 15;
  const int rb = blockIdx.x * 16;

  v8f accR[8], accI[8];
#pragma unroll
  for (int i = 0; i < 8; ++i) {
    accR[i] = vzero8();
    accI[i] = vzero8();
  }

  const size_t arow = (size_t)(rb + lr) * F_;
  for (int k = 0; k < F_; k += 4) {
    v2f aR = *(const v2f*)(Ar + arow + k + 2 * half);
    v2f aI = *(const v2f*)(Ai + arow + k + 2 * half);
    v2f aIn = -aI;
    __builtin_prefetch(Ar + arow + k + 32, 0, 3);
    __builtin_prefetch(Ai + arow + k + 32, 0, 3);
#pragma unroll
    for (int i = 0; i < 8; ++i) {
      const int n = (wv * 8 + i) * 16 + lr;
      const float2 w0 = W[(size_t)(k + 2 * half) * F_ + n];
      const float2 w1 = W[(size_t)(k + 2 * half + 1) * F_ + n];
      v2f br = {w0.x, w1.x};
      v2f bi = {w0.y, w1.y};
      accR[i] = wmma4(aR, br, accR[i]);
      accR[i] = wmma4(aIn, bi, accR[i]);
      accI[i] = wmma4(aR, bi, accI[i]);
      accI[i] = wmma4(aI, br, accI[i]);
    }
  }

#pragma unroll
  for (int j = 0; j < 8; ++j) {
    const int m = j + 8 * half;
    const size_t orow = (size_t)(rb + m) * F_;
#pragma unroll
    for (int i = 0; i < 8; ++i) {
      const int n = (wv * 8 + i) * 16 + lr;
      float2 o;
      o.x = accR[i][j];
      o.y = accI[i][j];
      out[orow + n] = o;
    }
  }
}

// ---------------------------------------------------------------------------
extern "C" void kernel_launch(void* const* d_in, const int* in_sizes, int n_in,
                              void* d_out, int out_size, void* d_ws,
                              size_t ws_size, hipStream_t stream) {
  (void)in_sizes;
  (void)n_in;
  (void)out_size;
  (void)ws_size;

  const float* kv = (const float*)d_in[0];
  const float* q = (const float*)d_in[1];
  const float* mask = (const float*)d_in[2];
  const float2* Wq = (const float2*)d_in[3];
  const float2* Wk = (const float2*)d_in[4];
  const float2* Wv = (const float2*)d_in[5];
  const float2* Wo = (const float2*)d_in[6];
  const float* lnq_s = (const float*)d_in[7];
  const float* lnq_b = (const float*)d_in[8];
  const float* lnk_s = (const float*)d_in[9];
  const float* lnk_b = (const float*)d_in[10];
  const float* lnv_s = (const float*)d_in[11];
  const float* lnv_b = (const float*)d_in[12];

  float* ws = (float*)d_ws;
  const size_t P = (size_t)B_ * L_ * F_;  // 4M elements per plane
  float* Qr = ws;
  float* Qi = ws + P;
  float* Kr = ws + 2 * P;
  float* Ki = ws + 3 * P;
  float* Vr = ws + 4 * P;
  float* Vi = ws + 5 * P;
  float* Or_ = ws + 6 * P;
  float* Oi_ = ws + 7 * P;

  const int MROWS = B_ * L_;  // 4096
  dim3 blk(256);

  proj_ln_kernel<<<MROWS / 16, blk, 0, stream>>>(q, Wq, lnq_s, lnq_b, Qr, Qi);
  proj_ln_kernel<<<MROWS / 16, blk, 0, stream>>>(kv, Wk, lnk_s, lnk_b, Kr, Ki);
  proj_ln_kernel<<<MROWS / 16, blk, 0, stream>>>(kv, Wv, lnv_s, lnv_b, Vr, Vi);

  const size_t shbytes = (size_t)ATTN_LDS_FLOATS * sizeof(float);  // 272,512
  (void)hipFuncSetAttribute((const void*)attn_kernel,
                            hipFuncAttributeMaxDynamicSharedMemorySize,
                            (int)shbytes);
  attn_kernel<<<B_ * H_ * (L_ / 16), blk, shbytes, stream>>>(
      Qr, Qi, Kr, Ki, Vr, Vi, mask, Or_, Oi_);

  oproj_kernel<<<MROWS / 16, blk, 0, stream>>>(Or_, Oi_, Wo, (float2*)d_out);
}